// LIF_R_40166534152624
// MI455X (gfx1250) — compile-verified
//
#include <hip/hip_runtime.h>

// ---------------------------------------------------------------------------
// LIF_R step, N=8192.  Dominant cost: I = w @ g  (8192x8192 fp32 GEMV,
// 256 MB of w -> pure HBM-bandwidth bound, ~11 us floor at 23.3 TB/s).
// GEMV via V_WMMA_F32_16X16X4_F32: per wave, A = 16x4 tile of w (16 output
// rows), B = g[k..k+3] broadcast over 16 columns, C = fp32 accumulator.
// Two independent accumulator chains hide the matrix-pipe dependent-issue
// latency.  2D grid (x = row blocks, y = K chunks) -> no integer division.
// Partials land in d_ws; a tiny second kernel fuses the LIF pointwise math.
// ---------------------------------------------------------------------------

typedef __attribute__((ext_vector_type(2))) float v2f;
typedef __attribute__((ext_vector_type(8))) float v8f;

#define NN 8192
#define ROWS_PER_WAVE 16
#define WAVES_PER_BLOCK 8       // 256 threads = 8 wave32
#define MAX_KCHUNKS 8

// LIF constants (from reference)
#define R_I       18.0f
#define F_V       0.12f
#define DELTA_V   12.0f

__global__ __launch_bounds__(256) void gemv_wmma_f32(
    const float* __restrict__ w,
    const float* __restrict__ g,
    float* __restrict__ partial,     // [kChunks][NN]
    int kPer)                        // columns handled per wave (NN / kChunks)
{
    const int lane     = threadIdx.x & 31;
    const int wave     = threadIdx.x >> 5;
    const int rowBlock = blockIdx.x * WAVES_PER_BLOCK + wave;  // 0 .. NN/16-1
    const int kChunk   = blockIdx.y;
    const int r0       = rowBlock * ROWS_PER_WAVE;
    const int k0       = kChunk * kPer;

    // A-fragment addressing (32-bit A 16x4 layout):
    //   lanes 0-15 : M = lane,    K = {0,1}
    //   lanes 16-31: M = lane-16, K = {2,3}
    const int m  = lane & 15;
    const int kh = lane >> 4;                    // 0 or 1 -> K offset 0 / 2

    const float* __restrict__ wp = w + (size_t)(r0 + m) * NN + k0 + 2 * kh;
    // B-fragment (32-bit B 4x16 layout): lane n holds column n; rows K split
    // {0,1} / {2,3} across lane halves.  B[k][n] = g[k0+k] (broadcast in n).
    const float* __restrict__ gp = g + k0 + 2 * kh;

    v8f c0 = {0.f, 0.f, 0.f, 0.f, 0.f, 0.f, 0.f, 0.f};
    v8f c1 = {0.f, 0.f, 0.f, 0.f, 0.f, 0.f, 0.f, 0.f};

    // Two K-blocks (8 columns) per trip; independent accumulator chains.
    const int trips = kPer >> 3;                 // kPer >= 1024, always even
    for (int it = 0; it < trips; ++it) {
        v2f a0 = *(const v2f*)(wp);
        v2f b0 = *(const v2f*)(gp);
        v2f a1 = *(const v2f*)(wp + 4);
        v2f b1 = *(const v2f*)(gp + 4);
        // keep the 16 sequential row streams ahead of the consumer
        __builtin_prefetch(wp + 1024, 0, 1);     // -> global_prefetch_b8
        // 8 args: (neg_a, A, neg_b, B, c_mod, C, reuse_a, reuse_b)
        c0 = __builtin_amdgcn_wmma_f32_16x16x4_f32(
                 false, a0, false, b0, (short)0, c0, false, false);
        c1 = __builtin_amdgcn_wmma_f32_16x16x4_f32(
                 false, a1, false, b1, (short)0, c1, false, false);
        wp += 8;
        gp += 8;
    }
    v8f c = c0 + c1;

    // D[m][n] == y_partial[r0+m] for every n.  C/D layout: VGPR j, lanes 0-15
    // hold (M=j, N=lane), lanes 16-31 hold (M=j+8, N=lane-16).  So at N=0:
    //   lane 0  : c[0..7] = y[r0+0 .. r0+7]
    //   lane 16 : c[0..7] = y[r0+8 .. r0+15]
    if ((lane & 15) == 0) {
        float* dst = partial + (size_t)kChunk * NN + r0 + (lane >> 4) * 8;
        *(v8f*)dst = c;
    }
}

__global__ __launch_bounds__(256) void lif_pointwise(
    const float* __restrict__ x_in,
    const float* __restrict__ v,
    const float* __restrict__ theta_s,
    const float* __restrict__ E_L,
    const float* __restrict__ C_m,
    const float* __restrict__ G,
    const float* __restrict__ partial,
    float* __restrict__ out,           // [0..NN) = v_new, [NN..2NN) = spiked_soft
    int kChunks)
{
    const int i = blockIdx.x * blockDim.x + threadIdx.x;
    if (i >= NN) return;

    float acc = x_in[i];
    for (int cidx = 0; cidx < kChunks; ++cidx)
        acc += partial[(size_t)cidx * NN + i];

    const float vi = v[i];
    const float th = theta_s[i];
    const float el = E_L[i];

    const float dv      = (G[i] * (el - vi) + acc * R_I) / C_m[i];
    const float v_next  = vi + dv;
    const float sig     = 1.0f / (1.0f + __expf(-(v_next - th)));  // surrogate
    const float spiked  = (v_next >= th) ? 1.0f : 0.0f;
    const float v_reset = el + F_V * (vi - el) - DELTA_V;
    const float v_new   = spiked * v_reset + (1.0f - spiked) * v_next;

    out[i]      = v_new;
    out[NN + i] = sig;
}

extern "C" void kernel_launch(void* const* d_in, const int* in_sizes, int n_in,
                              void* d_out, int out_size, void* d_ws, size_t ws_size,
                              hipStream_t stream)
{
    // setup_inputs order: x_in, v, g, theta_s, w, E_L, C_m, G, tau_g
    const float* x_in    = (const float*)d_in[0];
    const float* v       = (const float*)d_in[1];
    const float* g       = (const float*)d_in[2];
    const float* theta_s = (const float*)d_in[3];
    const float* w       = (const float*)d_in[4];
    const float* E_L     = (const float*)d_in[5];
    const float* C_m     = (const float*)d_in[6];
    const float* G       = (const float*)d_in[7];
    // d_in[8] = tau_g: unused (g_new is discarded hidden state)

    float* out     = (float*)d_out;
    float* partial = (float*)d_ws;

    // Shrink the K split if workspace is tight (power of two, deterministic).
    int kChunks = MAX_KCHUNKS;
    while (kChunks > 1 && (size_t)kChunks * NN * sizeof(float) > ws_size)
        kChunks >>= 1;
    const int kPer = NN / kChunks;

    const dim3 gemvGrid(NN / ROWS_PER_WAVE / WAVES_PER_BLOCK, kChunks);  // (64, kChunks)
    gemv_wmma_f32<<<gemvGrid, dim3(256), 0, stream>>>(w, g, partial, kPer);

    lif_pointwise<<<dim3((NN + 255) / 256), dim3(256), 0, stream>>>(
        x_in, v, theta_s, E_L, C_m, G, partial, out, kChunks);
}